// GATDecoder_11802570129772
// MI455X (gfx1250) — compile-verified
//
#include <hip/hip_runtime.h>
#include <math.h>

// Problem constants (match reference)
#define BATCH 32
#define DIM   2048
#define NNODE 80
#define NHEAD 4
#define CDIM  64
#define NEDGE 6320
#define HC    256   // NHEAD*CDIM

typedef __attribute__((ext_vector_type(2))) float v2f;
typedef __attribute__((ext_vector_type(8))) float v8f;

__device__ __forceinline__ float lrelu(float x) { return x > 0.f ? x : 0.2f * x; }

// ---------------------------------------------------------------------------
// K1: row-normalize image features: img_n[b] = img[b] / max(||img[b]||, 1e-12)
// ---------------------------------------------------------------------------
__global__ void k_normalize(const float* __restrict__ img, float* __restrict__ img_n) {
    int b = blockIdx.x;
    int t = threadIdx.x;
    __shared__ float red[8];
    float ss = 0.f;
    for (int j = t; j < DIM; j += 256) { float v = img[b * DIM + j]; ss += v * v; }
    for (int off = 16; off > 0; off >>= 1) ss += __shfl_xor(ss, off, 32);
    int lane = t & 31, wv = t >> 5;
    if (lane == 0) red[wv] = ss;
    __syncthreads();
    if (t == 0) { float s = 0.f; for (int i = 0; i < 8; ++i) s += red[i]; red[0] = s; }
    __syncthreads();
    float inv = 1.0f / fmaxf(sqrtf(red[0]), 1e-12f);
    for (int j = t; j < DIM; j += 256) img_n[b * DIM + j] = img[b * DIM + j] * inv;
}

// ---------------------------------------------------------------------------
// K2: generic fp32 WMMA GEMM: out[M,Nout] = A[M,K] @ W[K,Nout] + bias (opt relu)
// One wave computes one 16x16 tile via V_WMMA_F32_16X16X4_F32 over K.
// A-matrix per ISA 7.12.2: lanes0-15 M=0..15 K={0,1}; lanes16-31 K={2,3}.
// B mirrors with N across lanes. C/D: vgpr r -> row r (+8 for upper half-wave).
// ---------------------------------------------------------------------------
__global__ void __launch_bounds__(128)
k_gemm_wmma(const float* __restrict__ A, const float* __restrict__ W,
            const float* __restrict__ bias, float* __restrict__ out,
            int M, int K, int Nout, int relu) {
    int tid  = threadIdx.x;
    int lane = tid & 31;
    int wave = tid >> 5;
    int tile = blockIdx.x * (blockDim.x >> 5) + wave;
    int ntn  = Nout >> 4;
    int total = (M >> 4) * ntn;
    if (tile >= total) return;                  // uniform per wave -> EXEC stays all-1s
    int mt = tile / ntn;
    int nt = tile % ntn;
    int half = lane >> 4;                       // 0: K 0,1  1: K 2,3
    int l16  = lane & 15;
    int rowA = (mt << 4) + l16;
    int col  = (nt << 4) + l16;
    const float* Arow = A + (size_t)rowA * K + (half << 1);
    const float* Wcol = W + col + (size_t)(half << 1) * Nout;
    v8f acc = {};
    for (int k0 = 0; k0 < K; k0 += 4) {
        v2f a = *(const v2f*)(Arow + k0);       // 8B-aligned: b64 load
        v2f bm;
        bm.x = Wcol[(size_t)k0 * Nout];
        bm.y = Wcol[(size_t)(k0 + 1) * Nout];
        acc = __builtin_amdgcn_wmma_f32_16x16x4_f32(
            false, a, false, bm, (short)0, acc, false, false);
    }
    float bv = bias[col];
    #pragma unroll
    for (int r = 0; r < 8; ++r) {
        int row = (mt << 4) + r + (half << 3);
        float v = acc[r] + bv;
        if (relu) v = fmaxf(v, 0.f);
        out[(size_t)row * Nout + col] = v;
    }
}

// ---------------------------------------------------------------------------
// K3: deterministic CSR (bucket edges by dst). Single block.
// ---------------------------------------------------------------------------
__global__ void k_csr(const int* __restrict__ ei, int* __restrict__ offs,
                      int* __restrict__ perm) {
    __shared__ int cnt[NNODE];
    __shared__ int offsh[NNODE + 1];
    int t = threadIdx.x;
    if (t < NNODE) cnt[t] = 0;
    __syncthreads();
    const int* dst = ei + NEDGE;
    for (int e = t; e < NEDGE; e += blockDim.x) atomicAdd(&cnt[dst[e]], 1);
    __syncthreads();
    if (t == 0) {
        int run = 0;
        for (int n = 0; n < NNODE; ++n) { offsh[n] = run; run += cnt[n]; cnt[n] = offsh[n]; }
        offsh[NNODE] = run;
        for (int e = 0; e < NEDGE; ++e) { int d = dst[e]; perm[cnt[d]++] = e; }  // stable
    }
    __syncthreads();
    if (t < NNODE + 1) offs[t] = offsh[t];
}

// ---------------------------------------------------------------------------
// K4: materialize layer-1 xl/xr from the rank-1 decomposition:
//     xl[b,n,:] = (img_n@Wl1_img + bl1)[b,:] + Wl1[n,:]   (label one-hot rows)
// ---------------------------------------------------------------------------
__global__ void k_build1(const float* __restrict__ xlimg, const float* __restrict__ xrimg,
                         const float* __restrict__ Wl1, const float* __restrict__ Wr1,
                         float* __restrict__ xl, float* __restrict__ xr) {
    int bn = blockIdx.x;            // b*NNODE + n
    int hc = threadIdx.x;
    int b = bn / NNODE, n = bn % NNODE;
    size_t o = (size_t)bn * HC + hc;
    xl[o] = xlimg[b * HC + hc] + Wl1[(size_t)n * HC + hc];
    xr[o] = xrimg[b * HC + hc] + Wr1[(size_t)n * HC + hc];
}

// ---------------------------------------------------------------------------
// K5: edge logits. One wave per (edge,batch). Lane covers 8 channels;
// lanes 8h..8h+7 own head h; shuffle-reduce within groups of 8.
// logits[e,b,h] = sum_c att[h,c] * leaky(xl[b,src]+xr[b,dst]+w_e*We)
// ---------------------------------------------------------------------------
__global__ void k_edge_logits(const float* __restrict__ xl, const float* __restrict__ xr,
                              const float* __restrict__ We, const float* __restrict__ att,
                              const float* __restrict__ ew, const int* __restrict__ ei,
                              float* __restrict__ logits) {
    int gw   = (int)((blockIdx.x * blockDim.x + threadIdx.x) >> 5);
    int lane = threadIdx.x & 31;
    if (gw >= NEDGE * BATCH) return;
    int e = gw / BATCH, b = gw % BATCH;
    int s = ei[e], d = ei[NEDGE + e];
    int hc0 = lane * 8;
    const float4* xls = (const float4*)(xl + ((size_t)b * NNODE + s) * HC + hc0);
    const float4* xrd = (const float4*)(xr + ((size_t)b * NNODE + d) * HC + hc0);
    const float4* Wep = (const float4*)(We + hc0);
    const float4* atp = (const float4*)(att + hc0);
    float w = ew[e];
    float part = 0.f;
    #pragma unroll
    for (int q = 0; q < 2; ++q) {
        float4 a = xls[q], r = xrd[q], wv = Wep[q], at = atp[q];
        part += at.x * lrelu(a.x + r.x + w * wv.x);
        part += at.y * lrelu(a.y + r.y + w * wv.y);
        part += at.z * lrelu(a.z + r.z + w * wv.z);
        part += at.w * lrelu(a.w + r.w + w * wv.w);
    }
    for (int off = 1; off < 8; off <<= 1) part += __shfl_xor(part, off, 32);
    if ((lane & 7) == 0)
        logits[(size_t)(e * BATCH + b) * NHEAD + (lane >> 3)] = part;
}

// ---------------------------------------------------------------------------
// K6: per-(node,batch) softmax stats over incoming edges: mx & den [N,B,H]
// ---------------------------------------------------------------------------
__global__ void k_stats(const float* __restrict__ logits, const int* __restrict__ offs,
                        const int* __restrict__ perm, float* __restrict__ mx,
                        float* __restrict__ den) {
    int gw   = (int)((blockIdx.x * blockDim.x + threadIdx.x) >> 5);
    int lane = threadIdx.x & 31;
    if (gw >= NNODE * BATCH) return;
    int n = gw / BATCH, b = gw % BATCH;
    int beg = offs[n], end = offs[n + 1];
    float4 m = make_float4(-INFINITY, -INFINITY, -INFINITY, -INFINITY);
    for (int i = beg + lane; i < end; i += 32) {
        int e = perm[i];
        float4 lg = *(const float4*)(logits + (size_t)(e * BATCH + b) * NHEAD);
        m.x = fmaxf(m.x, lg.x); m.y = fmaxf(m.y, lg.y);
        m.z = fmaxf(m.z, lg.z); m.w = fmaxf(m.w, lg.w);
    }
    for (int off = 16; off > 0; off >>= 1) {
        m.x = fmaxf(m.x, __shfl_xor(m.x, off, 32));
        m.y = fmaxf(m.y, __shfl_xor(m.y, off, 32));
        m.z = fmaxf(m.z, __shfl_xor(m.z, off, 32));
        m.w = fmaxf(m.w, __shfl_xor(m.w, off, 32));
    }
    float4 s = make_float4(0.f, 0.f, 0.f, 0.f);
    for (int i = beg + lane; i < end; i += 32) {
        int e = perm[i];
        float4 lg = *(const float4*)(logits + (size_t)(e * BATCH + b) * NHEAD);
        s.x += __expf(lg.x - m.x); s.y += __expf(lg.y - m.y);
        s.z += __expf(lg.z - m.z); s.w += __expf(lg.w - m.w);
    }
    for (int off = 16; off > 0; off >>= 1) {
        s.x += __shfl_xor(s.x, off, 32); s.y += __shfl_xor(s.y, off, 32);
        s.z += __shfl_xor(s.z, off, 32); s.w += __shfl_xor(s.w, off, 32);
    }
    if (lane == 0) {
        *(float4*)(mx  + (size_t)gw * NHEAD) = m;
        *(float4*)(den + (size_t)gw * NHEAD) = s;
    }
}

// ---------------------------------------------------------------------------
// K7: aggregation. Block per (node,batch); thread per channel.
// out[b,n,hc] = [resid +] relu( sum_e alpha * xl[b,src_e,hc] + bias[hc] )
// ---------------------------------------------------------------------------
__global__ void k_aggregate(const float* __restrict__ xl, const float* __restrict__ logits,
                            const float* __restrict__ mx, const float* __restrict__ den,
                            const int* __restrict__ offs, const int* __restrict__ perm,
                            const int* __restrict__ ei, const float* __restrict__ bias,
                            const float* __restrict__ resid, float* __restrict__ out) {
    int nb = blockIdx.x;            // n*BATCH + b (matches stats layout)
    int n = nb / BATCH, b = nb % BATCH;
    int hc = threadIdx.x;
    int h = hc >> 6;
    float mxv = mx[(size_t)nb * NHEAD + h];
    float dn  = den[(size_t)nb * NHEAD + h];
    float invd = dn > 0.f ? 1.0f / dn : 0.f;
    int beg = offs[n], end = offs[n + 1];
    float acc = 0.f;
    for (int i = beg; i < end; ++i) {
        int e = perm[i];
        int s = ei[e];
        float alpha = __expf(logits[(size_t)(e * BATCH + b) * NHEAD + h] - mxv) * invd;
        acc += alpha * xl[((size_t)b * NNODE + s) * HC + hc];
    }
    float v = fmaxf(acc + bias[hc], 0.f);
    if (resid) v += resid[((size_t)b * NNODE + n) * HC + hc];
    out[((size_t)b * NNODE + n) * HC + hc] = v;
}

// ---------------------------------------------------------------------------
// K8: build MLP input: comb[bn, 0:256]=x2r, comb[bn, 256:384]=proj[b]
// ---------------------------------------------------------------------------
__global__ void k_comb(const float* __restrict__ x2, const float* __restrict__ proj,
                       float* __restrict__ comb) {
    int bn = blockIdx.x;
    int j  = threadIdx.x;           // 384
    int b  = bn / NNODE;
    comb[(size_t)bn * 384 + j] = (j < HC) ? x2[(size_t)bn * HC + j]
                                          : proj[b * 128 + (j - HC)];
}

// ---------------------------------------------------------------------------
// K9: head: out[bn] = h[bn,:] . Wm2 + bm2 (wave per row)
// ---------------------------------------------------------------------------
__global__ void k_head(const float* __restrict__ h, const float* __restrict__ Wm2,
                       const float* __restrict__ bm2, float* __restrict__ out) {
    int gw   = (int)((blockIdx.x * blockDim.x + threadIdx.x) >> 5);
    int lane = threadIdx.x & 31;
    if (gw >= BATCH * NNODE) return;
    float p = 0.f;
    for (int k = lane; k < 128; k += 32) p += h[(size_t)gw * 128 + k] * Wm2[k];
    for (int off = 16; off > 0; off >>= 1) p += __shfl_xor(p, off, 32);
    if (lane == 0) out[gw] = p + bm2[0];
}

// ---------------------------------------------------------------------------
extern "C" void kernel_launch(void* const* d_in, const int* in_sizes, int n_in,
                              void* d_out, int out_size, void* d_ws, size_t ws_size,
                              hipStream_t stream) {
    const float* img   = (const float*)d_in[0];
    const int*   ei    = (const int*)  d_in[1];   // [2,E]
    const float* ew    = (const float*)d_in[2];
    const float* Wl1   = (const float*)d_in[3];
    const float* bl1   = (const float*)d_in[4];
    const float* Wr1   = (const float*)d_in[5];
    const float* br1   = (const float*)d_in[6];
    const float* We1   = (const float*)d_in[7];
    const float* att1  = (const float*)d_in[8];
    const float* bias1 = (const float*)d_in[9];
    const float* Wl2   = (const float*)d_in[10];
    const float* bl2   = (const float*)d_in[11];
    const float* Wr2   = (const float*)d_in[12];
    const float* br2   = (const float*)d_in[13];
    const float* We2   = (const float*)d_in[14];
    const float* att2  = (const float*)d_in[15];
    const float* bias2 = (const float*)d_in[16];
    const float* Wproj = (const float*)d_in[17];
    const float* bproj = (const float*)d_in[18];
    const float* Wm1   = (const float*)d_in[19];
    const float* bm1   = (const float*)d_in[20];
    const float* Wm2   = (const float*)d_in[21];
    const float* bm2   = (const float*)d_in[22];
    float* out = (float*)d_out;

    // workspace carve-up (floats)
    float* ws = (float*)d_ws;
    size_t o = 0;
    float* img_n = ws + o; o += (size_t)BATCH * DIM;
    float* xlimg = ws + o; o += (size_t)BATCH * HC;
    float* xrimg = ws + o; o += (size_t)BATCH * HC;
    float* proj  = ws + o; o += (size_t)BATCH * 128;
    float* xlbuf = ws + o; o += (size_t)BATCH * NNODE * HC;   // xl1 then xl2
    float* xrbuf = ws + o; o += (size_t)BATCH * NNODE * HC;   // xr1 then xr2
    float* logit = ws + o; o += (size_t)NEDGE * BATCH * NHEAD;
    float* mxb   = ws + o; o += (size_t)NNODE * BATCH * NHEAD;
    float* denb  = ws + o; o += (size_t)NNODE * BATCH * NHEAD;
    float* x1    = ws + o; o += (size_t)BATCH * NNODE * HC;
    float* x2r   = ws + o; o += (size_t)BATCH * NNODE * HC;
    float* comb  = ws + o; o += (size_t)BATCH * NNODE * 384;
    float* hbuf  = ws + o; o += (size_t)BATCH * NNODE * 128;
    int* offs = (int*)(ws + o);
    int* perm = offs + (NNODE + 1);

    // 1. normalize
    k_normalize<<<BATCH, 256, 0, stream>>>(img, img_n);

    // 2. img-side projections (rank-1 decomposition of layer-1 linears) + proj
    //    Wl1 img rows start at row NNODE
    {
        int tiles = (BATCH / 16) * (HC / 16);     // 32
        k_gemm_wmma<<<tiles / 4, 128, 0, stream>>>(img_n, Wl1 + (size_t)NNODE * HC,
                                                   bl1, xlimg, BATCH, DIM, HC, 0);
        k_gemm_wmma<<<tiles / 4, 128, 0, stream>>>(img_n, Wr1 + (size_t)NNODE * HC,
                                                   br1, xrimg, BATCH, DIM, HC, 0);
        int tp = (BATCH / 16) * (128 / 16);       // 16
        k_gemm_wmma<<<tp / 4, 128, 0, stream>>>(img_n, Wproj, bproj, proj,
                                                BATCH, DIM, 128, 1);
    }

    // 3. CSR by dst
    k_csr<<<1, 256, 0, stream>>>(ei, offs, perm);

    // 4. layer-1 xl/xr
    k_build1<<<BATCH * NNODE, HC, 0, stream>>>(xlimg, xrimg, Wl1, Wr1, xlbuf, xrbuf);

    // 5-7. layer-1 edge softmax + aggregation -> x1
    k_edge_logits<<<(NEDGE * BATCH) / 8, 256, 0, stream>>>(xlbuf, xrbuf, We1, att1,
                                                           ew, ei, logit);
    k_stats<<<(NNODE * BATCH) / 8, 256, 0, stream>>>(logit, offs, perm, mxb, denb);
    k_aggregate<<<NNODE * BATCH, HC, 0, stream>>>(xlbuf, logit, mxb, denb, offs, perm,
                                                  ei, bias1, nullptr, x1);

    // 8. layer-2 linears (WMMA): x1 viewed as [B*N, 256]
    {
        int M = BATCH * NNODE;                    // 2560
        int tiles = (M / 16) * (HC / 16);         // 2560
        k_gemm_wmma<<<tiles / 4, 128, 0, stream>>>(x1, Wl2, bl2, xlbuf, M, HC, HC, 0);
        k_gemm_wmma<<<tiles / 4, 128, 0, stream>>>(x1, Wr2, br2, xrbuf, M, HC, HC, 0);
    }

    // 9-11. layer-2 edge softmax + aggregation (+residual x1) -> x2r
    k_edge_logits<<<(NEDGE * BATCH) / 8, 256, 0, stream>>>(xlbuf, xrbuf, We2, att2,
                                                           ew, ei, logit);
    k_stats<<<(NNODE * BATCH) / 8, 256, 0, stream>>>(logit, offs, perm, mxb, denb);
    k_aggregate<<<NNODE * BATCH, HC, 0, stream>>>(xlbuf, logit, mxb, denb, offs, perm,
                                                  ei, bias2, x1, x2r);

    // 12. concat [x2r | proj] and MLP layer 1 (WMMA)
    k_comb<<<BATCH * NNODE, 384, 0, stream>>>(x2r, proj, comb);
    {
        int M = BATCH * NNODE;
        int tiles = (M / 16) * (128 / 16);        // 1280
        k_gemm_wmma<<<tiles / 4, 128, 0, stream>>>(comb, Wm1, bm1, hbuf, M, 384, 128, 1);
    }

    // 13. head
    k_head<<<(BATCH * NNODE) / 8, 256, 0, stream>>>(hbuf, Wm2, bm2, out);
}